// Lin2d_89721866813987
// MI455X (gfx1250) — compile-verified
//
#include <hip/hip_runtime.h>

typedef __attribute__((ext_vector_type(2))) float v2f;
typedef __attribute__((ext_vector_type(8))) float v8f;

#define N_REPEATS 1000

// Kernel 1: compute T = (M^T)^1000 with sequential fp32 2x2 products
// (matches the reference's left-to-right iteration). One thread; trivial cost.
// T written row-major to d_ws: [T00, T01, T10, T11], so that y = x @ T.
__global__ void lin2d_power_kernel(const float* __restrict__ M,
                                   float* __restrict__ T) {
    if (threadIdx.x == 0 && blockIdx.x == 0) {
        // M row-major: M[0]=M00 M[1]=M01 M[2]=M10 M[3]=M11 ; tM = M^T
        const float tm00 = M[0], tm01 = M[2];
        const float tm10 = M[1], tm11 = M[3];
        float a00 = 1.f, a01 = 0.f, a10 = 0.f, a11 = 1.f;   // T = I
        for (int i = 0; i < N_REPEATS; ++i) {               // T = T @ tM
            float n00 = a00 * tm00 + a01 * tm10;
            float n01 = a00 * tm01 + a01 * tm11;
            float n10 = a10 * tm00 + a11 * tm10;
            float n11 = a10 * tm01 + a11 * tm11;
            a00 = n00; a01 = n01; a10 = n10; a11 = n11;
        }
        T[0] = a00; T[1] = a01; T[2] = a10; T[3] = a11;
    }
}

// Kernel 2: y = x @ T via V_WMMA_F32_16X16X4_F32.
// Per wave-tile of 32 pairs: D(16x16) = A(16x4) @ B(4x16), with
//   A = block-diag(T^T, T^T) at rows/cols {0,1} and {2,3} (rest zero):
//     lane 0/18: a = (T00, T10); lane 1/19: a = (T01, T11)   [A layout: V0=K0|K2, V1=K1|K3]
//   B = X^T for 32 pairs: lane l loads pair (32t + l); b = (x, y)
//     [B layout: V0 = K0 (lanes 0-15) | K2 (lanes 16-31); V1 = K1 | K3]
//   D rows 0..1 = group-1 results, rows 2..3 = group-2 results (lanes 0-15).
// Two shuffles route group-2 to lanes 16-31 -> one fully coalesced b64 store.
__global__ void lin2d_apply_wmma_kernel(const float* __restrict__ x,
                                        const float* __restrict__ T,
                                        float* __restrict__ y,
                                        int nTiles) {
    const int lane          = threadIdx.x & 31;
    const int wavesPerBlock = blockDim.x >> 5;
    const int wave          = blockIdx.x * wavesPerBlock + (threadIdx.x >> 5);
    const int nWaves        = gridDim.x * wavesPerBlock;

    const float T00 = T[0], T01 = T[1], T10 = T[2], T11 = T[3];

    v2f A = {0.f, 0.f};
    if (lane == 0 || lane == 18) { A.x = T00; A.y = T10; }  // (T^T) row of K0/K1 (or K2/K3)
    if (lane == 1 || lane == 19) { A.x = T01; A.y = T11; }

    const float2* __restrict__ xin  = reinterpret_cast<const float2*>(x);
    float2* __restrict__       yout = reinterpret_cast<float2*>(y);

    for (int t = wave; t < nTiles; t += nWaves) {           // wave-uniform branch: EXEC all 1s
        const int pairIdx = (t << 5) + lane;
        const float2 xv = xin[pairIdx];                     // coalesced global_load_b64 (256 B/wave)
        v2f Bv = {xv.x, xv.y};
        v8f c = {};
        // 8 args: (neg_a, A, neg_b, B, c_mod, C, reuse_a, reuse_b)
        c = __builtin_amdgcn_wmma_f32_16x16x4_f32(
                /*neg_a=*/false, A, /*neg_b=*/false, Bv,
                /*c_mod=*/(short)0, c, /*reuse_a=*/false, /*reuse_b=*/false);
        // Group-2 results (D rows 2,3) live in lanes 0-15; shuffle to lanes 16-31.
        const float g2x = __shfl(c[2], lane & 15, 32);
        const float g2y = __shfl(c[3], lane & 15, 32);
        float2 o;
        o.x = (lane < 16) ? c[0] : g2x;
        o.y = (lane < 16) ? c[1] : g2y;
        yout[pairIdx] = o;                                  // coalesced global_store_b64 (256 B/wave)
    }
}

// Scalar tail for nPairs % 32 (keeps WMMA kernel at full EXEC).
__global__ void lin2d_apply_tail_kernel(const float* __restrict__ x,
                                        const float* __restrict__ T,
                                        float* __restrict__ y,
                                        int startPair, int nPairs) {
    const int i = startPair + blockIdx.x * blockDim.x + threadIdx.x;
    if (i < nPairs) {
        const float2 xv = reinterpret_cast<const float2*>(x)[i];
        const float T00 = T[0], T01 = T[1], T10 = T[2], T11 = T[3];
        float2 o;
        o.x = xv.x * T00 + xv.y * T10;
        o.y = xv.x * T01 + xv.y * T11;
        reinterpret_cast<float2*>(y)[i] = o;
    }
}

extern "C" void kernel_launch(void* const* d_in, const int* in_sizes, int n_in,
                              void* d_out, int out_size, void* d_ws, size_t ws_size,
                              hipStream_t stream) {
    const float* x = (const float*)d_in[0];   // [B, 2] fp32
    const float* M = (const float*)d_in[1];   // [2, 2] fp32
    float*       y = (float*)d_out;           // [B, 2] fp32
    float*       T = (float*)d_ws;            // 4 floats scratch

    const int nPairs = in_sizes[0] / 2;
    const int nTiles = nPairs / 32;
    const int rem    = nPairs - nTiles * 32;

    lin2d_power_kernel<<<1, 32, 0, stream>>>(M, T);

    if (nTiles > 0) {
        // 256 threads = 8 waves/block; cap grid, grid-stride over tiles.
        int blocks = (nTiles + 7) / 8;
        if (blocks > 2048) blocks = 2048;
        lin2d_apply_wmma_kernel<<<blocks, 256, 0, stream>>>(x, T, y, nTiles);
    }
    if (rem > 0) {
        lin2d_apply_tail_kernel<<<1, 256, 0, stream>>>(x, T, y, nTiles * 32, nPairs);
    }
}